// NCC_VXM_59863254172029
// MI455X (gfx1250) — compile-verified
//
#include <hip/hip_runtime.h>
#include <hip/hip_bf16.h>

// NCC (local normalized cross-correlation), 9x9x9 window, vol [2,1,192,192,192] f32.
// Strategy (memory-bound, ~0.84 GB total traffic => ~36us at 23.3 TB/s):
//   K1: z-axis running-window sliding sums of {I,J,I2,J2,IJ} -> 5-ch scratch (283 MB)
//   K2: per-(b,d)-slice 16x16 tiles, H- then W- 9-tap sums done as f32 WMMA
//       (16x16x4) against a constant banded 0/1 matrix, fused cc + wave reduce
//   K3: deterministic tree reduction of per-wave partials -> -mean(cc)
// ws requirement: (5*14155776 + 55296) * 4 bytes ~= 283.3 MB.

#define PLANE   (192 * 192)
#define NBVOL   ((size_t)2 * 192 * 192 * 192)   // 14,155,776 voxels
#define WINSZ   729.0f
#define NTILES  (2 * 192 * 12 * 12)             // 55,296 16x16 tiles
#define TPW     3400                            // floats of LDS per wave: 5*600 + 400

typedef __attribute__((ext_vector_type(2))) float v2f;
typedef __attribute__((ext_vector_type(8))) float v8f;

// ---------------- Kernel 1: z-axis sliding sums (running window) ----------------
// One thread per (b,h,w-pair) column; float2 loads/stores, fully coalesced in w.
__global__ __launch_bounds__(256) void ncc_zpass(const float2* __restrict__ I2p,
                                                 const float2* __restrict__ J2p,
                                                 float2* __restrict__ zs) {
    int col = blockIdx.x * blockDim.x + threadIdx.x;   // exactly 2*192*96 threads
    int w2 = col % 96;
    int h  = (col / 96) % 192;
    int b  = col / (96 * 192);
    const size_t PL2 = PLANE / 2;
    const size_t NB2 = NBVOL / 2;
    size_t base = (size_t)b * 192 * PL2 + (size_t)h * 96 + w2;

    float2 sI = {0.f, 0.f}, sJ = {0.f, 0.f}, sI2 = {0.f, 0.f},
           sJ2 = {0.f, 0.f}, sIJ = {0.f, 0.f};
#pragma unroll
    for (int d = 0; d < 4; ++d) {
        float2 i = I2p[base + (size_t)d * PL2];
        float2 j = J2p[base + (size_t)d * PL2];
        sI.x += i.x;  sI.y += i.y;  sJ.x += j.x;  sJ.y += j.y;
        sI2.x += i.x * i.x; sI2.y += i.y * i.y;
        sJ2.x += j.x * j.x; sJ2.y += j.y * j.y;
        sIJ.x += i.x * j.x; sIJ.y += i.y * j.y;
    }
    for (int d = 0; d < 192; ++d) {
        int de = d + 4;
        if (de < 192) {
            float2 i = I2p[base + (size_t)de * PL2];
            float2 j = J2p[base + (size_t)de * PL2];
            sI.x += i.x;  sI.y += i.y;  sJ.x += j.x;  sJ.y += j.y;
            sI2.x += i.x * i.x; sI2.y += i.y * i.y;
            sJ2.x += j.x * j.x; sJ2.y += j.y * j.y;
            sIJ.x += i.x * j.x; sIJ.y += i.y * j.y;
            if (de + 8 < 192) {   // pull upcoming planes toward L2 (global_prefetch_b8)
                __builtin_prefetch(&I2p[base + (size_t)(de + 8) * PL2], 0, 1);
                __builtin_prefetch(&J2p[base + (size_t)(de + 8) * PL2], 0, 1);
            }
        }
        size_t o = base + (size_t)d * PL2;
        zs[0 * NB2 + o] = sI;
        zs[1 * NB2 + o] = sJ;
        zs[2 * NB2 + o] = sI2;
        zs[3 * NB2 + o] = sJ2;
        zs[4 * NB2 + o] = sIJ;
        int dl = d - 4;
        if (dl >= 0) {   // leaving plane: re-read (L2 hit, lag window ~2.6 MB)
            float2 i = I2p[base + (size_t)dl * PL2];
            float2 j = J2p[base + (size_t)dl * PL2];
            sI.x -= i.x;  sI.y -= i.y;  sJ.x -= j.x;  sJ.y -= j.y;
            sI2.x -= i.x * i.x; sI2.y -= i.y * i.y;
            sJ2.x -= j.x * j.x; sJ2.y -= j.y * j.y;
            sIJ.x -= i.x * j.x; sIJ.y -= i.y * j.y;
        }
    }
}

// --------- Kernel 2: H & W sliding sums via banded f32 WMMA + cc + reduce ---------
__global__ __launch_bounds__(128) void ncc_hwpass(const float* __restrict__ zs,
                                                  float* __restrict__ partials) {
    __shared__ float lds[4 * TPW];
    const int lane = threadIdx.x & 31;
    const int wave = threadIdx.x >> 5;
    const int gid  = blockIdx.x * 4 + wave;        // grid sized exactly: gid < NTILES
    const int tw    = gid % 12;
    const int th    = (gid / 12) % 12;
    const int slice = gid / 144;
    const int d     = slice % 192;
    const int b     = slice / 192;

    float* tile = &lds[wave * TPW];   // 5 channels x (24 rows x stride 25)
    float* out1 = tile + 5 * 600;     // 16 rows x stride 25 (H-pass staging)

    const size_t slicebase = ((size_t)b * 192 + d) * (size_t)PLANE;
    const int h0 = th * 16 - 4;
    const int w0 = tw * 16 - 4;       // divisible by 4 -> float4-aligned quads

    // Stage 24x24 halo tile per channel into LDS (zero-pad outside the volume).
    // Each row = six 16B-aligned float4 quads; every quad is entirely in or out.
    for (int ch = 0; ch < 5; ++ch) {
        const float* src = zs + (size_t)ch * NBVOL + slicebase;
        for (int idx = lane; idx < 24 * 6; idx += 32) {
            int r = idx / 6, q = idx % 6;
            int gh = h0 + r;
            int gw = w0 + 4 * q;
            float4 v = make_float4(0.f, 0.f, 0.f, 0.f);
            if (gh >= 0 && gh < 192 && gw >= 0 && gw < 192)
                v = *reinterpret_cast<const float4*>(src + (size_t)gh * 192 + gw);
            float* dst = tile + ch * 600 + r * 25 + 4 * q;
            dst[0] = v.x; dst[1] = v.y; dst[2] = v.z; dst[3] = v.w;
        }
    }
    // Each wave owns its LDS region; intra-wave ds ordering suffices (no barrier).

    const int m  = lane & 15;          // A-frag row / B-frag column (N)
    const int kh = (lane >> 4) * 2;    // K offset of this lane's pair within a K=4 chunk

    // Constant banded 0/1 fragment, shared by the H-pass A operand and the
    // W-pass B operand (band[c].v = 1 iff (4c+kh+v) in [m, m+8]). Hoisted once.
    v2f band[6];
#pragma unroll
    for (int c6 = 0; c6 < 6; ++c6) {
        const int k0 = 4 * c6 + kh;
        band[c6].x = (k0     >= m && k0     <= m + 8) ? 1.f : 0.f;
        band[c6].y = (k0 + 1 >= m && k0 + 1 <= m + 8) ? 1.f : 0.f;
    }

    v8f res[5];
    for (int ch = 0; ch < 5; ++ch) {
        const float* t = tile + ch * 600;

        // H pass: out1(16x24) = BandH(16x24) * tile(24x24); BandH[m][k]=1 iff m<=k<=m+8
        for (int nt = 0; nt < 2; ++nt) {
            const int n0   = nt * 16;
            const int ncol = n0 + m;
            const bool cok = (ncol < 24);
            v8f acc = {};
#pragma unroll
            for (int c6 = 0; c6 < 6; ++c6) {
                const int k0 = 4 * c6 + kh;
                v2f bb;
                bb.x = cok ? t[(k0    ) * 25 + ncol] : 0.f;
                bb.y = cok ? t[(k0 + 1) * 25 + ncol] : 0.f;
                acc = __builtin_amdgcn_wmma_f32_16x16x4_f32(
                        false, band[c6], false, bb, (short)0, acc, false, false);
            }
            // D layout: lane L, VGPR r -> M = r + 8*(L/16), N = L%16
#pragma unroll
            for (int r = 0; r < 8; ++r) {
                int M = r + 8 * (lane >> 4);
                if (cok) out1[M * 25 + ncol] = acc[r];
            }
        }

        // W pass: out2(16x16) = out1(16x24) * BandW(24x16); BandW[k][n]=1 iff n<=k<=n+8
        v8f acc2 = {};
#pragma unroll
        for (int c6 = 0; c6 < 6; ++c6) {
            const int k0 = 4 * c6 + kh;
            v2f a;
            a.x = out1[m * 25 + k0];
            a.y = out1[m * 25 + k0 + 1];
            acc2 = __builtin_amdgcn_wmma_f32_16x16x4_f32(
                     false, a, false, band[c6], (short)0, acc2, false, false);
        }
        res[ch] = acc2;
    }

    // Fused cc formula on the 8 voxels each lane owns, then wave32 reduce.
    float local = 0.f;
#pragma unroll
    for (int r = 0; r < 8; ++r) {
        float Is = res[0][r], Js = res[1][r];
        float I2 = res[2][r], J2 = res[3][r], IJ = res[4][r];
        float uI = Is * (1.0f / WINSZ);
        float uJ = Js * (1.0f / WINSZ);
        float cross = IJ - uJ * Is - uI * Js + uI * uJ * WINSZ;
        float Iv = I2 - 2.0f * uI * Is + uI * uI * WINSZ;
        float Jv = J2 - 2.0f * uJ * Js + uJ * uJ * WINSZ;
        local += cross * cross / (Iv * Jv + 1e-5f);
    }
#pragma unroll
    for (int off = 16; off > 0; off >>= 1)
        local += __shfl_down(local, off, 32);
    if (lane == 0) partials[gid] = local;
}

// ---------------- Kernel 3: deterministic final reduction ----------------
__global__ __launch_bounds__(256) void ncc_finalize(const float* __restrict__ partials,
                                                    float* __restrict__ out) {
    __shared__ float sm[256];
    float s = 0.f;
    for (int i = threadIdx.x; i < NTILES; i += 256) s += partials[i];  // fixed order
    sm[threadIdx.x] = s;
    __syncthreads();
    for (int off = 128; off > 0; off >>= 1) {
        if (threadIdx.x < off) sm[threadIdx.x] += sm[threadIdx.x + off];
        __syncthreads();
    }
    if (threadIdx.x == 0) out[0] = -sm[0] / (float)NBVOL;
}

extern "C" void kernel_launch(void* const* d_in, const int* in_sizes, int n_in,
                              void* d_out, int out_size, void* d_ws, size_t ws_size,
                              hipStream_t stream) {
    const float2* I = (const float2*)d_in[0];   // y_true
    const float2* J = (const float2*)d_in[1];   // y_pred
    float* zs       = (float*)d_ws;             // 5 * NBVOL floats
    float* partials = zs + 5 * NBVOL;           // NTILES floats

    // K1: 2*192*96 float2-columns / 256 = 144 blocks (exact)
    ncc_zpass<<<144, 256, 0, stream>>>(I, J, (float2*)zs);
    // K2: NTILES waves, 4 waves per block of 128 threads (exact)
    ncc_hwpass<<<NTILES / 4, 128, 0, stream>>>(zs, partials);
    // K3: single-block deterministic reduce
    ncc_finalize<<<1, 256, 0, stream>>>(partials, (float*)d_out);
}